// RNN_38062000177373
// MI455X (gfx1250) — compile-verified
//
#include <hip/hip_runtime.h>

typedef __attribute__((ext_vector_type(2))) float v2f;
typedef __attribute__((ext_vector_type(8))) float v8f;

#define B_SIZE 256
#define T_SIZE 1024
#define D_SIZE 512
#define UNITS  2

// ---------------------------------------------------------------------------
// Phase 1: xwT[t][b][u] = bias[u] + sum_k x[b][t][k] * W[k][u]
// WMMA mapping (V_WMMA_F32_16X16X4_F32, wave32):
//   A (16x4)  = W^T zero-padded: A[m][k] = (m<2) ? W[k0+k][m] : 0
//   B (4x16)  = x tile: B[k][n] = x[b0+n][t][k0+k]
//   D (16x16) : D[m][n] = xw[b0+n][unit m]  -> VGPR0/1, lanes 0..15
// A-layout (ISA 7.12.2, 32-bit A 16x4): lane L<16 holds M=L, K=0..1 in
// VGPR0..1; lanes 16..31 hold K=2..3.  We pre-swizzle W into LDS so each
// lane fetches its A pair with one conflict-free ds_load_b64.
// ---------------------------------------------------------------------------
__global__ __launch_bounds__(256) void rnn_xproj_wmma(
    const float* __restrict__ x,    // [B, T, D]
    const float* __restrict__ W,    // [D, UNITS]
    const float* __restrict__ bias, // [UNITS]
    float* __restrict__ xwT)        // [T, B, UNITS] workspace
{
    __shared__ v2f sWA[128 * 32];   // 32 KB: [kstep 0..127][lane 0..31]

    const int tid = threadIdx.x;

    // Build the swizzled / zero-padded A image once per workgroup.
    for (int idx = tid; idx < 128 * 32; idx += 256) {
        const int ks   = idx >> 5;
        const int lane = idx & 31;
        const int m    = lane & 15;
        const int kb   = ks * 4 + ((lane >> 4) << 1);
        v2f a;
        a.x = (m < UNITS) ? W[(kb + 0) * UNITS + m] : 0.0f;
        a.y = (m < UNITS) ? W[(kb + 1) * UNITS + m] : 0.0f;
        sWA[idx] = a;
    }
    __syncthreads();

    const float bias0 = bias[0];
    const float bias1 = bias[1];

    const int lane    = tid & 31;
    const int wave    = tid >> 5;                 // 0..7
    const int wglob   = blockIdx.x * 8 + wave;    // 0..2047
    const int btile   = wglob >> 7;               // 0..15
    const int s       = wglob & 127;              // 0..127 -> t block of 8
    const int brow    = btile * 16 + (lane & 15); // batch row for this lane
    const int khalf   = (lane >> 4) << 1;         // 0 or 2 (K split at lane 16)

    const float* xrow = x + (size_t)brow * T_SIZE * D_SIZE + khalf;

    // 8 consecutive t per wave, processed in pairs: two independent WMMA
    // accumulator chains share each A-fragment load.
    for (int tp = 0; tp < 4; ++tp) {
        const int t0 = s * 8 + tp * 2;
        const float* p0 = xrow + (size_t)t0 * D_SIZE;
        const float* p1 = p0 + D_SIZE;

        v8f c0 = {bias0, bias1, 0.f, 0.f, 0.f, 0.f, 0.f, 0.f};
        v8f c1 = c0;

        #pragma unroll 4
        for (int ks = 0; ks < 128; ++ks) {
            const v2f a  = sWA[ks * 32 + lane];            // ds_load_b64
            const v2f x0 = *(const v2f*)(p0 + ks * 4);     // global b64
            const v2f x1 = *(const v2f*)(p1 + ks * 4);
            c0 = __builtin_amdgcn_wmma_f32_16x16x4_f32(
                     false, a, false, x0, (short)0, c0, false, false);
            c1 = __builtin_amdgcn_wmma_f32_16x16x4_f32(
                     false, a, false, x1, (short)0, c1, false, false);
        }

        // D[0][n] in c[0], D[1][n] in c[1] for lanes n = 0..15:
        // one fully coalesced 128B store per tile into [T,B,2].
        if (lane < 16) {
            v2f r0; r0.x = c0[0]; r0.y = c0[1];
            v2f r1; r1.x = c1[0]; r1.y = c1[1];
            *(v2f*)(xwT + ((size_t)(t0 + 0) * B_SIZE + brow) * UNITS) = r0;
            *(v2f*)(xwT + ((size_t)(t0 + 1) * B_SIZE + brow) * UNITS) = r1;
        }
    }
}

// ---------------------------------------------------------------------------
// Phase 2: sequential scan over T.  One lane per batch row (256 lanes),
// h kept in registers, xwT reads are coalesced (lane = b), hardware
// v_tanh_f32 keeps the dependent chain short (~2 FMA + 1 TRANS per step).
// ---------------------------------------------------------------------------
__global__ __launch_bounds__(256) void rnn_scan(
    const float* __restrict__ xwT, // [T, B, UNITS]
    const float* __restrict__ U,   // [UNITS, UNITS]
    float* __restrict__ out)       // [B, UNITS]
{
    const int b = threadIdx.x;     // single block of 256
    const float u00 = U[0], u01 = U[1], u10 = U[2], u11 = U[3];

    float h0 = 0.0f, h1 = 0.0f;
    const float* p = xwT + b * UNITS;

    #pragma unroll 8
    for (int t = 0; t < T_SIZE; ++t) {
        const v2f c = *(const v2f*)(p + (size_t)t * B_SIZE * UNITS);
        const float p0 = fmaf(h0, u00, fmaf(h1, u10, c.x));
        const float p1 = fmaf(h0, u01, fmaf(h1, u11, c.y));
        // CDNA5 hardware tanh (TRANS op).  Two back-to-back tanh: the second
        // is the "independent op" for the first; v_nops cover the second
        // before its result is consumed (TRANS hazard rule, ISA 5 §7.4).
        asm volatile("v_tanh_f32 %0, %2\n\t"
                     "v_tanh_f32 %1, %3\n\t"
                     "v_nop\n\t"
                     "v_nop"
                     : "=v"(h0), "=v"(h1)
                     : "v"(p0), "v"(p1));
    }

    v2f r; r.x = h0; r.y = h1;
    *(v2f*)(out + b * UNITS) = r;
}

// ---------------------------------------------------------------------------
extern "C" void kernel_launch(void* const* d_in, const int* in_sizes, int n_in,
                              void* d_out, int out_size, void* d_ws, size_t ws_size,
                              hipStream_t stream) {
    const float* x    = (const float*)d_in[0]; // [B,T,D]
    const float* W    = (const float*)d_in[1]; // [D,UNITS]
    const float* U    = (const float*)d_in[2]; // [UNITS,UNITS]
    const float* bias = (const float*)d_in[3]; // [UNITS]
    float* out = (float*)d_out;                // [B,UNITS]
    float* xwT = (float*)d_ws;                 // needs T*B*UNITS*4 = 2 MB

    // 16 b-tiles x 128 t-blocks = 2048 waves = 256 blocks x 8 waves.
    rnn_xproj_wmma<<<256, 256, 0, stream>>>(x, W, bias, xwT);
    rnn_scan<<<1, 256, 0, stream>>>(xwT, U, out);
}